// LaminiIndex_24343874634160
// MI455X (gfx1250) — compile-verified
//
#include <hip/hip_runtime.h>
#include <math.h>
#include <stdint.h>

// Problem constants (from reference setup_inputs)
#define B_DIM   4
#define L_DIM   512
#define D_DIM   384
#define N_DIM   16384
#define BL      (B_DIM * L_DIM)        // 2048 rows
#define TOPK    16

// Kernel-1 tiling: 128-row M-tile per block, 8 waves x 16 rows; all waves
// share the same 16 key-columns per pass (8x reuse of B through LDS).
#define M_TILE     128
#define PASS_COLS  16
#define CHUNK_N    2048                // columns per block (grid.y)
#define NPASS      (CHUNK_N / PASS_COLS)   // 128
#define NCHUNK     (N_DIM / CHUNK_N)       // 8

// TDM-padded LDS layout: rows of 384 dwords, pad 4 dwords every 128 dwords
// (pad_interval=6, pad_amount=3) -> row stride 396 words, segment stride 132.
#define LSTRIDE    396
#define SEG_W      132                 // 128 data + 4 pad words
#define SS         21                  // score row stride (coprime with 64)

// LDS byte offsets (total 296960 B < 320 KB WGP LDS)
#define Q_OFF      0
#define Q_BYTES    (M_TILE * LSTRIDE * 4)        // 202752
#define B_OFF      (Q_OFF + Q_BYTES)
#define BBUF_BYTES (PASS_COLS * LSTRIDE * 4)     // 25344
#define B_BYTES    (2 * BBUF_BYTES)              // 50688
#define S_OFF      (B_OFF + B_BYTES)
#define S_BYTES    (M_TILE * SS * 4)             // 10752
#define CV_OFF     (S_OFF + S_BYTES)
#define CV_BYTES   (M_TILE * 32 * 4)             // 16384
#define CI_OFF     (CV_OFF + CV_BYTES)
#define CI_BYTES   (M_TILE * 32 * 4)
#define SMEM_TOTAL (CI_OFF + CI_BYTES)           // 296960

typedef __attribute__((ext_vector_type(2))) float        v2f;
typedef __attribute__((ext_vector_type(8))) float        v8f;
typedef __attribute__((ext_vector_type(4))) unsigned int v4u;
typedef __attribute__((ext_vector_type(8))) int          v8i;
typedef __attribute__((ext_vector_type(4))) int          v4i;

// ---------------------------------------------------------------------------
// TDM 2D tile load: global (row-major, D_DIM elements/row, 4B elements)
// -> LDS at lds_off, padded +4 dwords every 128 dwords (bank-conflict-free
// fragment reads at stride 396). D# per CDNA5 ISA §8.3/8.4.
// This toolchain's builtin is the 6-arg form:
//   (u32x4 g0, i32x8 g1, i32x4 g2, i32x4 g3, i32x8 g4, i32 cpol)
// ---------------------------------------------------------------------------
__device__ __forceinline__ void tdm_load_2d(unsigned lds_off, const void* gptr,
                                            unsigned tile_d1,
                                            unsigned tensor_d1) {
  unsigned long long ga = (unsigned long long)(uintptr_t)gptr;
  v4u g0 = {
      1u,                                               // count=1, no gather
      lds_off,                                          // lds_addr (bytes)
      (unsigned)(ga & 0xffffffffu),                     // global_addr[31:0]
      (unsigned)((ga >> 32) & 0x01ffffffu) | 0x80000000u // addr[56:32]|type=2
  };
  v8i g1 = {
      (int)((2u << 16) | (1u << 20) | (6u << 22) | (3u << 25)),
      //     data=4B     pad_en       int=128dw    amt=4dw ; wg_mask=0
      (int)((D_DIM & 0xffffu) << 16),                   // tensor_dim0 lo16
      (int)((tensor_d1 & 0xffffu) << 16),               // dim0 hi=0 | dim1 lo16
      (int)(((tensor_d1 >> 16) & 0xffffu) | ((unsigned)D_DIM << 16)), // tile_dim0
      (int)(tile_d1 & 0xffffu),                         // tile_dim1 | tile_dim2=0
      (int)D_DIM,                                       // tensor_dim0_stride lo32
      0, 0
  };
  v4i g2 = {0, 0, 0, 0};
  v4i g3 = {0, 0, 0, 0};
  v8i g4 = {0, 0, 0, 0, 0, 0, 0, 0};
  __builtin_amdgcn_tensor_load_to_lds(g0, g1, g2, g3, g4, 0);
}

// ---------------------------------------------------------------------------
// Kernel 1: scores = q @ keys^T + gumbel(u); per-(128-row tile, chunk) top-16.
// TDM stages q (once) and double-buffered 16-column key tiles into LDS;
// 8 waves each run a 16x16 f32 WMMA tile per pass (96 chained MMAs, K=384).
// ---------------------------------------------------------------------------
__global__ __launch_bounds__(256) void lamini_score_topk(
    const float* __restrict__ q, const float* __restrict__ keys,
    const float* __restrict__ gn, float* __restrict__ ws_val,
    int* __restrict__ ws_idx) {
  extern __shared__ char smem[];
  float* qtile  = (float*)(smem + Q_OFF);
  float* btile  = (float*)(smem + B_OFF);
  float* scores = (float*)(smem + S_OFF);
  float* cval   = (float*)(smem + CV_OFF);
  int*   cidx   = (int*)(smem + CI_OFF);

  const int tid       = threadIdx.x;
  const int rowBase   = blockIdx.x * M_TILE;
  const int chunkBase = blockIdx.y * CHUNK_N;

  const int wave  = tid >> 5;
  const int lane  = tid & 31;
  const int lhalf = lane >> 4;       // 0 or 1
  const int l16   = lane & 15;
  const int kb    = lhalf * 2;       // K sub-offset of A/B fragments

  const int srow = tid >> 1;         // scan row 0..127
  const int ssub = tid & 1;          // 2 scan threads/row, 8 cols each/pass

  // Prologue: TDM-stage q tile (128x384) and first B tile (16x384).
  if (wave == 0) {
    tdm_load_2d(Q_OFF, q + (size_t)rowBase * D_DIM, M_TILE, BL);
    tdm_load_2d(B_OFF, keys + (size_t)chunkBase * D_DIM, PASS_COLS, N_DIM);
    __builtin_amdgcn_s_wait_tensorcnt(0);
  }
  __syncthreads();

  // Per-thread running top-16 (sorted descending; tv[15] = current min).
  float tv[TOPK]; int ti[TOPK];
  #pragma unroll
  for (int s = 0; s < TOPK; ++s) { tv[s] = -INFINITY; ti[s] = 0; }

  for (int p = 0; p < NPASS; ++p) {
    const int colBase = chunkBase + p * PASS_COLS;
    const int ncol    = colBase + l16;

    // Double-buffer: issue next B tile, then ensure current one landed.
    if (wave == 0) {
      if (p + 1 < NPASS) {
        tdm_load_2d(B_OFF + (unsigned)(((p + 1) & 1) * BBUF_BYTES),
                    keys + (size_t)(colBase + PASS_COLS) * D_DIM,
                    PASS_COLS, N_DIM);
        __builtin_amdgcn_s_wait_tensorcnt(1);  // oldest (pass p) complete
      } else {
        __builtin_amdgcn_s_wait_tensorcnt(0);
      }
    }
    __syncthreads();

    // A: 16x4 f32 from qtile (lane m = l16 within wave's 16-row subtile)
    // B: 4x16 f32 from btile (lane n = l16), both in TDM-padded layout.
    v8f c = {0.f, 0.f, 0.f, 0.f, 0.f, 0.f, 0.f, 0.f};
    const float* abase = qtile + (wave * 16 + l16) * LSTRIDE + kb;
    const float* bbase = btile + (size_t)((p & 1) * (BBUF_BYTES / 4))
                               + l16 * LSTRIDE + kb;
    #pragma unroll
    for (int seg = 0; seg < 3; ++seg) {
      const float* as = abase + seg * SEG_W;
      const float* bs = bbase + seg * SEG_W;
      #pragma unroll 4
      for (int dd = 0; dd < 128; dd += 4) {
        v2f a = *(const v2f*)(as + dd);
        v2f b = *(const v2f*)(bs + dd);
        c = __builtin_amdgcn_wmma_f32_16x16x4_f32(
            /*neg_a=*/false, a, /*neg_b=*/false, b,
            /*c_mod=*/(short)0, c, /*reuse_a=*/false, /*reuse_b=*/false);
      }
    }

    // Add gumbel g = -log(-log(u)) (tau rescales only; rank-invariant).
    // C layout: VGPR r -> M = lhalf*8 + r, N = l16.
    #pragma unroll
    for (int r = 0; r < 8; ++r) {
      int m    = wave * 16 + lhalf * 8 + r;
      float u  = gn[(size_t)(rowBase + m) * N_DIM + ncol];
      float sv = c[r] - __logf(-__logf(u));
      scores[m * SS + l16] = sv;
    }
    __syncthreads();

    // Running top-16: thread (srow, ssub) scans its 8 columns of this pass.
    #pragma unroll
    for (int j = 0; j < 8; ++j) {
      int col  = ssub * 8 + j;
      float v  = scores[srow * SS + col];
      int gidx = colBase + col;
      if (v > tv[TOPK - 1]) {
        int pos = TOPK - 1;
        #pragma unroll
        for (int s = TOPK - 2; s >= 0; --s) if (v > tv[s]) pos = s;
        #pragma unroll
        for (int s = TOPK - 1; s >= 1; --s)
          if (s > pos) { tv[s] = tv[s - 1]; ti[s] = ti[s - 1]; }
        tv[pos] = v; ti[pos] = gidx;
      }
    }
    __syncthreads();
  }

  // Dump 2x16 candidates per row, merge 32 -> 16, write partial lists.
  #pragma unroll
  for (int s = 0; s < TOPK; ++s) {
    cval[srow * 32 + ssub * TOPK + s] = tv[s];
    cidx[srow * 32 + ssub * TOPK + s] = ti[s];
  }
  __syncthreads();

  if (tid < M_TILE) {
    const int row = tid;
    float fv[TOPK]; int fi[TOPK];
    #pragma unroll
    for (int s = 0; s < TOPK; ++s) { fv[s] = -INFINITY; fi[s] = 0; }
    for (int c0 = 0; c0 < 32; ++c0) {
      float v = cval[row * 32 + c0];
      int  ix = cidx[row * 32 + c0];
      if (v > fv[TOPK - 1]) {
        int pos = TOPK - 1;
        #pragma unroll
        for (int s = TOPK - 2; s >= 0; --s) if (v > fv[s]) pos = s;
        #pragma unroll
        for (int s = TOPK - 1; s >= 1; --s)
          if (s > pos) { fv[s] = fv[s - 1]; fi[s] = fi[s - 1]; }
        fv[pos] = v; fi[pos] = ix;
      }
    }
    const size_t base = ((size_t)(rowBase + row) * NCHUNK + blockIdx.y) * TOPK;
    #pragma unroll
    for (int s = 0; s < TOPK; ++s) {
      ws_val[base + s] = fv[s];
      ws_idx[base + s] = fi[s];
    }
  }
}

// ---------------------------------------------------------------------------
// Kernel 2: per row, merge 8 partial top-16 lists (128 candidates) -> final
// top-16, then out = (1/16) * sum of gathered keys / values rows.
// (Forward value of the straight-through estimator == normalized hard mask.)
// ---------------------------------------------------------------------------
__global__ __launch_bounds__(256) void lamini_merge_gather(
    const float* __restrict__ keys, const float* __restrict__ values,
    const float* __restrict__ ws_val, const int* __restrict__ ws_idx,
    float* __restrict__ out) {
  __shared__ float cv[NCHUNK * TOPK];
  __shared__ int   ci[NCHUNK * TOPK];
  __shared__ int   fidx[TOPK];

  const int row = blockIdx.x;
  const int tid = threadIdx.x;

  if (tid < NCHUNK * TOPK) {
    cv[tid] = ws_val[(size_t)row * NCHUNK * TOPK + tid];
    ci[tid] = ws_idx[(size_t)row * NCHUNK * TOPK + tid];
  }
  __syncthreads();

  if (tid == 0) {
    float fv[TOPK]; int fi[TOPK];
    #pragma unroll
    for (int s = 0; s < TOPK; ++s) { fv[s] = -INFINITY; fi[s] = 0; }
    for (int c0 = 0; c0 < NCHUNK * TOPK; ++c0) {
      float v = cv[c0]; int ix = ci[c0];
      if (v > fv[TOPK - 1]) {
        int pos = TOPK - 1;
        #pragma unroll
        for (int s = TOPK - 2; s >= 0; --s) if (v > fv[s]) pos = s;
        #pragma unroll
        for (int s = TOPK - 1; s >= 1; --s)
          if (s > pos) { fv[s] = fv[s - 1]; fi[s] = fi[s - 1]; }
        fv[pos] = v; fi[pos] = ix;
      }
    }
    #pragma unroll
    for (int s = 0; s < TOPK; ++s) fidx[s] = fi[s];
  }
  __syncthreads();

  for (int d = tid; d < D_DIM; d += 256) {
    float ak = 0.f, av = 0.f;
    #pragma unroll
    for (int s = 0; s < TOPK; ++s) {
      int ix = fidx[s];
      ak += keys[(size_t)ix * D_DIM + d];
      av += values[(size_t)ix * D_DIM + d];
    }
    out[(size_t)row * D_DIM + d] = ak * (1.0f / TOPK);
    out[(size_t)BL * D_DIM + (size_t)row * D_DIM + d] = av * (1.0f / TOPK);
  }
}

// ---------------------------------------------------------------------------
extern "C" void kernel_launch(void* const* d_in, const int* in_sizes, int n_in,
                              void* d_out, int out_size, void* d_ws, size_t ws_size,
                              hipStream_t stream) {
  (void)in_sizes; (void)n_in; (void)out_size; (void)ws_size;
  const float* q      = (const float*)d_in[0];   // (B, L, D) f32
  const float* keys   = (const float*)d_in[1];   // (N, D)    f32
  const float* values = (const float*)d_in[2];   // (N, D)    f32
  const float* gn     = (const float*)d_in[3];   // (BL, N)   f32 uniforms

  // Workspace: partial top-16 lists (2048 rows * 8 chunks * 16): 2 MB.
  float* ws_val = (float*)d_ws;
  int*   ws_idx = (int*)((char*)d_ws + sizeof(float) * (size_t)BL * NCHUNK * TOPK);

  dim3 g1(BL / M_TILE, NCHUNK);                  // 16 x 8 workgroups
  lamini_score_topk<<<g1, 256, SMEM_TOTAL, stream>>>(q, keys, gn, ws_val, ws_idx);
  lamini_merge_gather<<<BL, 256, 0, stream>>>(keys, values, ws_val, ws_idx,
                                              (float*)d_out);
}